// Scalar_27590869909554
// MI455X (gfx1250) — compile-verified
//
#include <hip/hip_runtime.h>
#include <hip/hip_bf16.h>
#include <cstdint>

// ---------------- problem constants ----------------
#define NYV   256          // velocity grid
#define PMLW  20
#define NP    296          // padded wavefield dim (256 + 2*20)
#define GOFF  4            // zero guard band on each side
#define NS    328          // guarded stride (room for last tile halo)
#define NN    (NS*NS)      // one shot-plane
#define NSHOT 2
#define NSRC  2
#define NREC  256
#define NT    300
#define TILE  32
#define HTP   36           // TILE + 4  (psi tiles / extended psi region)
#define HTW   40           // TILE + 8  (wfc tile: halo for psi-halo recompute)
#define DTF   0.0004f
#define INV_DX   0.25f
#define INV_DX2  0.0625f

// D1 = (1/12, -8/12, 0, 8/12, -1/12); D2 = (-1/12, 16/12, -30/12, 16/12, -1/12)
#define C1A  (1.0f/12.0f)
#define C1B  (-8.0f/12.0f)
#define D2A  (-1.0f/12.0f)
#define D2B  (16.0f/12.0f)
#define D2C  (-30.0f/12.0f)

typedef uint32_t u32;
typedef u32 u32x4 __attribute__((ext_vector_type(4)));
typedef u32 u32x8 __attribute__((ext_vector_type(8)));

// ---- Tensor Data Mover: 2-D tile (td x td f32, row stride NS) global -> LDS ----
// D# group0/group1 per CDNA5 ISA §8.3/8.4. Issued once per wave (EXEC ignored),
// completion tracked with TENSORcnt.
template<int TD>
__device__ __forceinline__ void tdm_load_tile(const float* gbase, int row0, int col0, void* lds)
{
    unsigned long long ga = (unsigned long long)(uintptr_t)gbase
                          + 4ull * (unsigned long long)(row0 * NS + col0);
    u32x4 g0 = { 1u,                                   // count=1, user desc
                 (u32)(uintptr_t)lds,                  // LDS byte address
                 (u32)ga,                              // global_addr[31:0]
                 ((u32)(ga >> 32) & 0x01FFFFFFu) | 0x80000000u }; // addr[56:32] | type=2
    u32x8 g1 = { 0x00020000u,                          // workgroup_mask=0, data_size=2 (4B)
                 (u32)NS << 16,                        // tensor_dim0[15:0] @ bits 63:48
                 (u32)NS << 16,                        // tensor_dim0 hi=0; tensor_dim1[15:0]
                 (u32)TD << 16,                        // tensor_dim1 hi=0; tile_dim0=TD
                 (u32)TD,                              // tile_dim1=TD; tile_dim2=0
                 (u32)NS,                              // tensor_dim0_stride low32
                 0u, 0u };                             // stride hi / dim1_stride unused (2D)
    asm volatile("tensor_load_to_lds %0, %1" :: "s"(g0), "s"(g1) : "memory");
}

// ---------------- init kernels ----------------
__global__ void zero_kernel(float* p, int n)
{
    int i = blockIdx.x * blockDim.x + threadIdx.x;
    for (; i < n; i += gridDim.x * blockDim.x) p[i] = 0.0f;
}

__global__ void v2dt2_kernel(const float* __restrict__ v, float* __restrict__ v2)
{
    int j = blockIdx.x * blockDim.x + threadIdx.x;
    int i = blockIdx.y * blockDim.y + threadIdx.y;
    if (i >= NP || j >= NP) return;
    int iv = min(max(i - PMLW, 0), NYV - 1);
    int jv = min(max(j - PMLW, 0), NYV - 1);
    float vv = v[iv * NYV + jv];
    v2[(i + GOFF) * NS + (j + GOFF)] = vv * vv * (DTF * DTF);
}

__global__ void profiles_kernel(float* __restrict__ pa, float* __restrict__ pb)
{
    int i = blockIdx.x * blockDim.x + threadIdx.x;
    if (i >= NP) return;
    float x   = (float)i;
    float dlo = fminf(fmaxf((20.0f - x) * (1.0f / 20.0f), 0.0f), 1.0f);
    float dhi = fminf(fmaxf((x - 275.0f) * (1.0f / 20.0f), 0.0f), 1.0f);
    float d   = fmaxf(dlo, dhi);
    float sigma = 302.21429345546814f * d * d;       // 3*3500*ln(100)/(2*80)
    float alpha = 78.53981633974483f * (1.0f - d);   // pi*25*(1-d)
    float a = expf(-(sigma + alpha) * DTF);
    float b = sigma / (sigma + alpha + 1e-9f) * (a - 1.0f);
    bool in = d > 0.0f;
    pa[i] = in ? a : 0.0f;
    pb[i] = in ? b : 0.0f;
}

// ---------------- fused time step: psi (halo recompute) + zeta + leapfrog ----------------
// Reads: wfc, psix_in, psiy_in (TDM tiles), wfp, zetax, zetay, v2 (pointwise)
// Writes: psix_out, psiy_out (double-buffered), zetax, zetay, wfn (into wfp buffer)
__global__ void __launch_bounds__(256)
step_kernel(const float* __restrict__ wfc, float* __restrict__ wfp /* in: t-1, out: wfn */,
            const float* __restrict__ psix_in, const float* __restrict__ psiy_in,
            float* __restrict__ psix_out, float* __restrict__ psiy_out,
            float* __restrict__ zetax, float* __restrict__ zetay,
            const float* __restrict__ v2, const float* __restrict__ pa,
            const float* __restrict__ pb)
{
    __shared__ float sw [HTW * HTW];   // wfc tile,   field rows i0-4 .. i0+35
    __shared__ float spx[HTP * HTP];   // psix tile,  field rows i0-2 .. i0+33
    __shared__ float spy[HTP * HTP];   // psiy tile
    const int s  = blockIdx.z;
    const int i0 = blockIdx.y * TILE;
    const int j0 = blockIdx.x * TILE;
    const int tid = threadIdx.y * 32 + threadIdx.x;

    // --- parallel TDM tile loads: waves 0,1,2 issue, each waits its own TENSORcnt ---
    if (threadIdx.y == 0) {
        tdm_load_tile<HTW>(wfc     + s * NN, i0,     j0,     (void*)sw);
        __builtin_amdgcn_s_wait_tensorcnt(0);
    } else if (threadIdx.y == 1) {
        tdm_load_tile<HTP>(psix_in + s * NN, i0 + 2, j0 + 2, (void*)spx);
        __builtin_amdgcn_s_wait_tensorcnt(0);
    } else if (threadIdx.y == 2) {
        tdm_load_tile<HTP>(psiy_in + s * NN, i0 + 2, j0 + 2, (void*)spy);
        __builtin_amdgcn_s_wait_tensorcnt(0);
    }
    __syncthreads();

    // --- phase 1: update psi over the 36x36 extended region (in LDS, in place) ---
    for (int e = tid; e < HTP * HTP; e += 256) {
        int er = e / HTP, ec = e % HTP;
        int i = i0 + er - 2, j = j0 + ec - 2;
        float px = 0.0f, py = 0.0f;
        if (i >= 0 && i < NP && j >= 0 && j < NP) {
            // wfc tile index of field cell (i,j): row i-i0+4 = er+2, col ec+2
            float dwdx = INV_DX * (C1A * sw[(er    ) * HTW + ec + 2]
                                 + C1B * sw[(er + 1) * HTW + ec + 2]
                                 - C1B * sw[(er + 3) * HTW + ec + 2]
                                 - C1A * sw[(er + 4) * HTW + ec + 2]);
            float dwdy = INV_DX * (C1A * sw[(er + 2) * HTW + ec    ]
                                 + C1B * sw[(er + 2) * HTW + ec + 1]
                                 - C1B * sw[(er + 2) * HTW + ec + 3]
                                 - C1A * sw[(er + 2) * HTW + ec + 4]);
            px = pa[i] * spx[e] + pb[i] * dwdx;
            py = pa[j] * spy[e] + pb[j] * dwdy;
            if (er >= 2 && er < 2 + TILE && ec >= 2 && ec < 2 + TILE) {
                int idx = s * NN + (i + GOFF) * NS + (j + GOFF);
                psix_out[idx] = px;
                psiy_out[idx] = py;
            }
        }
        spx[e] = px;      // zero outside physical domain = reference zero padding
        spy[e] = py;
    }
    __syncthreads();

    // --- phase 2: zeta + leapfrog over the interior 32x32 ---
    const int tx = threadIdx.x;
#pragma unroll
    for (int rr = 0; rr < 4; ++rr) {
        int ry = threadIdx.y + rr * 8;
        int i = i0 + ry, j = j0 + tx;
        if (i < NP && j < NP) {
            int lw = ry + 4, lc = tx + 4;          // wfc tile coords of (i,j)
            int lp = ry + 2, pc = tx + 2;          // psi tile coords of (i,j)
            float d2x = INV_DX2 * (D2A * sw[(lw - 2) * HTW + lc] + D2B * sw[(lw - 1) * HTW + lc]
                                 + D2C * sw[lw * HTW + lc]
                                 + D2B * sw[(lw + 1) * HTW + lc] + D2A * sw[(lw + 2) * HTW + lc]);
            float d2y = INV_DX2 * (D2A * sw[lw * HTW + lc - 2] + D2B * sw[lw * HTW + lc - 1]
                                 + D2C * sw[lw * HTW + lc]
                                 + D2B * sw[lw * HTW + lc + 1] + D2A * sw[lw * HTW + lc + 2]);
            float dpx = INV_DX * (C1A * spx[(lp - 2) * HTP + pc] + C1B * spx[(lp - 1) * HTP + pc]
                                - C1B * spx[(lp + 1) * HTP + pc] - C1A * spx[(lp + 2) * HTP + pc]);
            float dpy = INV_DX * (C1A * spy[lp * HTP + pc - 2] + C1B * spy[lp * HTP + pc - 1]
                                - C1B * spy[lp * HTP + pc + 1] - C1A * spy[lp * HTP + pc + 2]);
            int idx  = s * NN + (i + GOFF) * NS + (j + GOFF);
            int vidx = (i + GOFF) * NS + (j + GOFF);
            float zx = pa[i] * zetax[idx] + pb[i] * (d2x + dpx);
            float zy = pa[j] * zetay[idx] + pb[j] * (d2y + dpy);
            zetax[idx] = zx;
            zetay[idx] = zy;
            float lap = d2x + d2y + dpx + dpy + zx + zy;
            float wc  = sw[lw * HTW + lc];
            wfp[idx] = fmaf(v2[vidx], lap, 2.0f * wc - wfp[idx]);
        }
    }
}

// ---------------- source injection + receiver gather ----------------
__global__ void __launch_bounds__(512)
src_rec_kernel(float* __restrict__ wfn, const float* __restrict__ v2,
               const float* __restrict__ amps, const int* __restrict__ sloc,
               const int* __restrict__ rloc, float* __restrict__ out, int t)
{
    int tid = threadIdx.x;
    if (tid < NSHOT * NSRC) {
        int s = tid / NSRC, k = tid % NSRC;
        int sy = sloc[(s * NSRC + k) * 2 + 0] + PMLW + GOFF;
        int sx = sloc[(s * NSRC + k) * 2 + 1] + PMLW + GOFF;
        float amp = amps[(s * NSRC + k) * NT + t];
        float sv  = v2[sy * NS + sx];
        atomicAdd(&wfn[s * NN + sy * NS + sx], sv * amp);
    }
    __threadfence();
    __syncthreads();
    int s = tid / NREC, r = tid % NREC;
    int ry = rloc[(s * NREC + r) * 2 + 0] + PMLW + GOFF;
    int rx = rloc[(s * NREC + r) * 2 + 1] + PMLW + GOFF;
    float val = __hip_atomic_load(&wfn[s * NN + ry * NS + rx],
                                  __ATOMIC_RELAXED, __HIP_MEMORY_SCOPE_AGENT);
    out[(s * NREC + r) * NT + t] = val;
}

// ---------------- host driver ----------------
extern "C" void kernel_launch(void* const* d_in, const int* in_sizes, int n_in,
                              void* d_out, int out_size, void* d_ws, size_t ws_size,
                              hipStream_t stream)
{
    const float* v    = (const float*)d_in[0];
    const float* amps = (const float*)d_in[1];
    const int*   sloc = (const int*)d_in[2];
    const int*   rloc = (const int*)d_in[3];
    float*       out  = (float*)d_out;

    const size_t SZ = (size_t)NSHOT * NN;   // one state array (all shots)
    float* ws     = (float*)d_ws;
    float* wfc    = ws + 0 * SZ;
    float* wfp    = ws + 1 * SZ;
    float* psix_a = ws + 2 * SZ;
    float* psix_b = ws + 3 * SZ;
    float* psiy_a = ws + 4 * SZ;
    float* psiy_b = ws + 5 * SZ;
    float* zetax  = ws + 6 * SZ;
    float* zetay  = ws + 7 * SZ;
    float* v2     = ws + 8 * SZ;            // NN floats (shot-independent)
    float* pa     = v2 + NN;                // NP floats
    float* pb     = pa + 320;               // NP floats
    const int total = (int)(8 * SZ + NN + 640);

    zero_kernel<<<2048, 256, 0, stream>>>(ws, total);
    {
        dim3 b(16, 16), g((NP + 15) / 16, (NP + 15) / 16);
        v2dt2_kernel<<<g, b, 0, stream>>>(v, v2);
    }
    profiles_kernel<<<(NP + 255) / 256, 256, 0, stream>>>(pa, pb);

    dim3 blk(32, 8);
    dim3 grd((NP + TILE - 1) / TILE, (NP + TILE - 1) / TILE, NSHOT);
    for (int t = 0; t < NT; ++t) {
        step_kernel<<<grd, blk, 0, stream>>>(wfc, wfp, psix_a, psiy_a, psix_b, psiy_b,
                                             zetax, zetay, v2, pa, pb);
        src_rec_kernel<<<1, 512, 0, stream>>>(wfp, v2, amps, sloc, rloc, out, t);
        float* tmp;
        tmp = wfc;    wfc    = wfp;    wfp    = tmp;   // (wfn, wfc) -> (wfc, wfp)
        tmp = psix_a; psix_a = psix_b; psix_b = tmp;   // psi double buffer
        tmp = psiy_a; psiy_a = psiy_b; psiy_b = tmp;
    }
}